// FlowEstimator_180388627249
// MI455X (gfx1250) — compile-verified
//
#include <hip/hip_runtime.h>

typedef __attribute__((ext_vector_type(2))) float v2f;
typedef __attribute__((ext_vector_type(8))) float v8f;

#define H_ 80
#define W_ 80
#define D_ 20
#define C_ 32
#define HWD_ (H_*W_*D_)          // 128000, channel stride
#define NDISP_ 405
#define MASK_OFF_ ((long)NDISP_*HWD_)

__device__ __forceinline__ float clip01(float x) {
  return fminf(fmaxf(x, 0.0f), 1.0f);
}

// depth-summed, clipped alpha mask at (h,w)
__device__ __forceinline__ float mask_sum(const float* __restrict__ a, int h, int w) {
  const float* p = a + ((long)h*W_ + w)*D_;
  float s = 0.0f;
#pragma unroll
  for (int q = 0; q < D_; ++q) s += p[q];
  return clip01(s);
}

__global__ void mask_precompute(const float* __restrict__ a1, const float* __restrict__ a2,
                                float* __restrict__ m1, float* __restrict__ m2) {
  int idx = blockIdx.x * blockDim.x + threadIdx.x;
  if (idx >= H_*W_) return;
  int h = idx / W_, w = idx % W_;
  m1[idx] = mask_sum(a1, h, w);
  m2[idx] = mask_sum(a2, h, w);
}

template<bool USE_WS>
__global__ __launch_bounds__(640)
void corr_wmma(const float* __restrict__ f1, const float* __restrict__ a1,
               const float* __restrict__ f2, const float* __restrict__ a2,
               const float* __restrict__ m1g, const float* __restrict__ m2g,
               float* __restrict__ out)
{
  const int lane = threadIdx.x & 31;
  const int d    = threadIdx.x >> 5;          // one wave per depth slice
  const int h    = blockIdx.x / 5;
  const int w0   = (blockIdx.x % 5) * 16;
  const int n    = lane & 15;                 // A: row M; B: col N
  const int hi   = lane >> 4;                 // K-half selector / C row bank

  // ---- A matrix (reused for all 405 displacements): A[M=n][K] = f1[K, h, w0+n, d]
  v2f A[8];
  {
    const float* ap = f1 + ((long)h*W_ + (w0 + n))*D_ + d;
#pragma unroll
    for (int t = 0; t < 8; ++t) {
      const int c0 = 4*t + 2*hi;
      A[t].x = ap[(long)c0 * HWD_];
      A[t].y = ap[(long)(c0 + 1) * HWD_];
    }
  }

  // ---- mask1 for the 8 output rows this lane owns (M = v + 8*hi)
  float m1v[8];
#pragma unroll
  for (int v = 0; v < 8; ++v) {
    const int w = w0 + v + 8*hi;
    m1v[v] = USE_WS ? m1g[h*W_ + w] : mask_sum(a1, h, w);
  }

  for (int k = 0; k < 5; ++k) {               // depth displacement index
    const int  dzd = d + k - 2;
    const bool din = (dzd >= 0) & (dzd < D_);
    for (int i = 0; i < 9; ++i) {             // row displacement index
      const int  hh  = h + i - 4;
      const bool hin = (hh >= 0) & (hh < H_);
#pragma unroll
      for (int tile = 0; tile < 2; ++tile) {  // two N-tiles cover q = w0-4 .. w0+19
        const int  q   = w0 + (tile ? 12 : -4) + n;
        const bool qin = (q >= 0) & (q < W_);

        // mask2 with pad=1.0 in any out-of-range dimension
        float m2v = 1.0f;
        if (din & hin & qin)
          m2v = USE_WS ? m2g[hh*W_ + q] : mask_sum(a2, hh, q);

        v8f acc = {0.f,0.f,0.f,0.f,0.f,0.f,0.f,0.f};
        if (din && hin) {                     // wave-uniform: EXEC all-ones for WMMA
          const int    qs = qin ? q : 0;      // safe address, value masked below
          const float* bp = f2 + ((long)hh*W_ + qs)*D_ + dzd;
          v2f B[8];
#pragma unroll
          for (int t = 0; t < 8; ++t) {
            const int c0 = 4*t + 2*hi;
            const float bx = bp[(long)c0 * HWD_];
            const float by = bp[(long)(c0 + 1) * HWD_];
            B[t].x = qin ? bx : 0.0f;
            B[t].y = qin ? by : 0.0f;
          }
#pragma unroll
          for (int t = 0; t < 8; ++t) {       // K = 32 via 8 chained fp32 WMMAs
            acc = __builtin_amdgcn_wmma_f32_16x16x4_f32(
                false, A[t], false, B[t], (short)0, acc, false, false);
          }
        }

        // scatter the valid dx band: dx = q_n - (w0 + m), |dx| <= 4
#pragma unroll
        for (int v = 0; v < 8; ++v) {
          const int m  = v + 8*hi;
          const int dx = tile ? (n + 12 - m) : (n - 4 - m);
          if (dx >= -4 && dx <= 4) {
            const int  Dix = k*81 + i*9 + (dx + 4);
            const long o   = (((long)Dix*H_ + h)*W_ + (w0 + m))*D_ + d;
            out[o]             = acc[v] * (1.0f/32.0f);   // mean over c=32
            out[MASK_OFF_ + o] = clip01(m1v[v] * m2v);
          }
        }
      }
    }
  }
}

extern "C" void kernel_launch(void* const* d_in, const int* in_sizes, int n_in,
                              void* d_out, int out_size, void* d_ws, size_t ws_size,
                              hipStream_t stream) {
  (void)in_sizes; (void)n_in; (void)out_size;
  const float* f1 = (const float*)d_in[0];
  const float* a1 = (const float*)d_in[1];
  const float* f2 = (const float*)d_in[2];
  const float* a2 = (const float*)d_in[3];
  float* out = (float*)d_out;

  if (d_ws && ws_size >= (size_t)(2 * H_ * W_ * sizeof(float))) {
    float* m1 = (float*)d_ws;
    float* m2 = m1 + H_*W_;
    mask_precompute<<<(H_*W_ + 255)/256, 256, 0, stream>>>(a1, a2, m1, m2);
    corr_wmma<true><<<400, 640, 0, stream>>>(f1, a1, f2, a2, m1, m2, out);
  } else {
    corr_wmma<false><<<400, 640, 0, stream>>>(f1, a1, f2, a2, nullptr, nullptr, out);
  }
}